// minGRUBi_80066780332440
// MI455X (gfx1250) — compile-verified
//
#include <hip/hip_runtime.h>

typedef __attribute__((ext_vector_type(2))) float v2f;
typedef __attribute__((ext_vector_type(8))) float v8f;

#define SEQ     4096
#define BATCH   4
#define DIM_    1024
#define DINNER  1536
#define NCOLS   3072   // 2*DINNER
#define NCHUNK  16
#define CHUNK   256    // SEQ / NCHUNK
#define MROWS   16384  // BATCH*SEQ
#define PITCH   20     // LDS row pitch (floats): conflict-free b64 frag reads, 16B-aligned rows

__device__ __forceinline__ float sigm(float x) { return 1.0f / (1.0f + __expf(-x)); }
__device__ __forceinline__ float g_fn(float x) { return x >= 0.0f ? x + 0.5f : sigm(x); }

// 32-bit LDS byte offset of a __shared__ object (AS3 pointers are 32-bit).
__device__ __forceinline__ unsigned lds_addr32(const void* p) {
  return (unsigned)(uintptr_t)(__attribute__((address_space(3))) const void*)p;
}
// CDNA5 async DMA: memory -> LDS, tracked by ASYNCcnt (no VGPR staging).
__device__ __forceinline__ void async_load_b128(unsigned lds, const void* gaddr) {
  asm volatile("global_load_async_to_lds_b128 %0, %1, off"
               :: "v"(lds), "v"(gaddr) : "memory");
}
__device__ __forceinline__ void wait_async0() {
  asm volatile("s_wait_asynccnt 0x0" ::: "memory");
}

// ---------------------------------------------------------------------------
// C[M][N] = A[M][K] * Bt[N][K]^T   (all row-major fp32)
// 128x128 tile / block (256 threads = 8 waves). Wave: 32x64 subtile = 2x4 WMMA
// accumulators of V_WMMA_F32_16X16X4_F32. LDS is [row][k] (pitch 20) so each
// fragment is one aligned ds_load_b64. Double-buffered via
// GLOBAL_LOAD_ASYNC_TO_LDS_B128: next slab streams into the other buffer
// while the matrix pipe runs -- zero staging VGPRs, no spills.
// ---------------------------------------------------------------------------
__global__ __launch_bounds__(256) void gemm_f32_wmma(const float* __restrict__ A,
                                                     const float* __restrict__ Bt,
                                                     float* __restrict__ C,
                                                     int M, int N, int K) {
  __shared__ float sA[2][128][PITCH];
  __shared__ float sB[2][128][PITCH];

  const int tid  = threadIdx.x;
  const int lane = tid & 31;
  const int wid  = tid >> 5;      // 0..7
  const int wm   = wid >> 1;      // 0..3 : 32-row slab
  const int wn   = wid & 1;       // 0..1 : 64-col slab
  const int half = lane >> 4;     // selects K pair (A/B) and +8 row (C/D)
  const int lm   = lane & 15;

  const int n0 = blockIdx.x * 128;
  const int m0 = blockIdx.y * 128;

  const int ldRow = tid >> 2;          // 0..63 (two passes cover 128 rows)
  const int ldK   = (tid & 3) << 2;    // 0,4,8,12

  const float* Abase = A + (size_t)m0 * K + ldK;
  const float* Bbase = Bt + (size_t)n0 * K + ldK;

  v8f acc[2][4] = {};

  // ---- prologue: async-stage slab k0=0 into buffer 0 ----
#pragma unroll
  for (int rr = 0; rr < 2; ++rr) {
    const int row = ldRow + rr * 64;
    async_load_b128(lds_addr32(&sA[0][row][ldK]), Abase + (size_t)row * K);
    async_load_b128(lds_addr32(&sB[0][row][ldK]), Bbase + (size_t)row * K);
  }
  wait_async0();
  __syncthreads();

  int cur = 0;
  for (int k0 = 0; k0 < K; k0 += 16) {
    const bool more = (k0 + 16) < K;

    // ---- stream next slab into the other buffer (overlaps WMMAs) ----
    if (more) {
      const int nxt = cur ^ 1;
#pragma unroll
      for (int rr = 0; rr < 2; ++rr) {
        const int row = ldRow + rr * 64;
        async_load_b128(lds_addr32(&sA[nxt][row][ldK]),
                        Abase + (size_t)row * K + k0 + 16);
        async_load_b128(lds_addr32(&sB[nxt][row][ldK]),
                        Bbase + (size_t)row * K + k0 + 16);
      }
      if (k0 + 32 < K) {  // near-cache global_prefetch_b8 for the slab after
        __builtin_prefetch(Abase + (size_t)ldRow * K + k0 + 32, 0, 3);
        __builtin_prefetch(Bbase + (size_t)ldRow * K + k0 + 32, 0, 3);
      }
    }

    // ---- 4 k-slices of 4 -> 32 WMMAs; fragments are single ds_load_b64 ----
#pragma unroll
    for (int kk = 0; kk < 4; ++kk) {
      const int kb = kk * 4 + 2 * half;
      v2f a[2], b[4];
#pragma unroll
      for (int i = 0; i < 2; ++i)
        a[i] = *reinterpret_cast<const v2f*>(&sA[cur][wm * 32 + i * 16 + lm][kb]);
#pragma unroll
      for (int j = 0; j < 4; ++j)
        b[j] = *reinterpret_cast<const v2f*>(&sB[cur][wn * 64 + j * 16 + lm][kb]);
#pragma unroll
      for (int i = 0; i < 2; ++i)
#pragma unroll
        for (int j = 0; j < 4; ++j)
          acc[i][j] = __builtin_amdgcn_wmma_f32_16x16x4_f32(
              false, a[i], false, b[j], (short)0, acc[i][j], false, false);
    }

    // ---- all waves: own asyncs done, then barrier -> buffer ready ----
    if (more) wait_async0();
    __syncthreads();
    cur ^= 1;
  }

  // ---- write back: C/D layout: VGPR r -> M = r + 8*half, lanes = N ----
#pragma unroll
  for (int i = 0; i < 2; ++i) {
#pragma unroll
    for (int j = 0; j < 4; ++j) {
      const int col = n0 + wn * 64 + j * 16 + lm;
#pragma unroll
      for (int r = 0; r < 8; ++r) {
        const int row = m0 + wm * 32 + i * 16 + half * 8 + r;
        C[(size_t)row * N + col] = acc[i][j][r];
      }
    }
  }
}

// ---------------------------------------------------------------------------
// Pass A: per-(dir, b, chunk, e) chunk summary: P = prod c, h = local scan
// tail.  h_t = c*h + v, c = sigmoid(-gate), v = sigmoid(gate)*g(hidden).
// grid: x = DINNER/256, y = NCHUNK, z = dir*BATCH + b
// ---------------------------------------------------------------------------
__global__ __launch_bounds__(256) void scan_partials(const float* __restrict__ hg,
                                                     float* __restrict__ sumC,
                                                     float* __restrict__ sumH) {
  const int e   = blockIdx.x * 256 + threadIdx.x;
  const int kc  = blockIdx.y;
  const int db  = blockIdx.z;          // dir*4 + b
  const int dir = db >> 2;
  const int b   = db & 3;

  float h = 0.0f, P = 1.0f;
  for (int t = 0; t < CHUNK; ++t) {
    const int p    = kc * CHUNK + t;
    const int sidx = dir ? (SEQ - 1 - p) : p;
    const size_t row = ((size_t)b * SEQ + sidx) * NCOLS;
    const float gt = hg[row + DINNER + e];
    const float hd = hg[row + e];
    const float c  = sigm(-gt);
    const float v  = (1.0f - c) * g_fn(hd);
    h = fmaf(c, h, v);
    P *= c;
  }
  const size_t o = ((size_t)db * NCHUNK + kc) * DINNER + e;
  sumC[o] = P;
  sumH[o] = h;
}

// ---------------------------------------------------------------------------
// Pass B: tiny sequential prefix over chunk summaries -> carry-in per chunk.
// ---------------------------------------------------------------------------
__global__ __launch_bounds__(256) void scan_carry(const float* __restrict__ sumC,
                                                  const float* __restrict__ sumH,
                                                  float* __restrict__ carry) {
  const int idx = blockIdx.x * 256 + threadIdx.x;  // 0..12287
  const int e   = idx % DINNER;
  const int db  = idx / DINNER;                    // 0..7
  float h = 0.0f;
  for (int k = 0; k < NCHUNK; ++k) {
    const size_t o = ((size_t)db * NCHUNK + k) * DINNER + e;
    carry[o] = h;
    h = fmaf(sumC[o], h, sumH[o]);
  }
}

// ---------------------------------------------------------------------------
// Pass C: replay both directions' chunks with carry-in; both directions land
// at the same output index p, so fuse fwd+bwd and store the sum once.
// grid: x = DINNER/256, y = NCHUNK, z = b
// ---------------------------------------------------------------------------
__global__ __launch_bounds__(256) void scan_apply(const float* __restrict__ hg,
                                                  const float* __restrict__ carry,
                                                  float* __restrict__ sb) {
  const int e  = blockIdx.x * 256 + threadIdx.x;
  const int kc = blockIdx.y;
  const int b  = blockIdx.z;

  float h0 = carry[((size_t)(b)     * NCHUNK + kc) * DINNER + e];  // forward
  float h1 = carry[((size_t)(4 + b) * NCHUNK + kc) * DINNER + e];  // backward

  for (int t = 0; t < CHUNK; ++t) {
    const int p = kc * CHUNK + t;
    {  // forward: reads original position p
      const size_t row = ((size_t)b * SEQ + p) * NCOLS;
      const float c = sigm(-hg[row + DINNER + e]);
      h0 = fmaf(c, h0, (1.0f - c) * g_fn(hg[row + e]));
    }
    {  // backward: reads original position SEQ-1-p, lands at output p
      const size_t row = ((size_t)b * SEQ + (SEQ - 1 - p)) * NCOLS;
      const float c = sigm(-hg[row + DINNER + e]);
      h1 = fmaf(c, h1, (1.0f - c) * g_fn(hg[row + e]));
    }
    sb[((size_t)b * SEQ + p) * DINNER + e] = h0 + h1;
  }
}

// ---------------------------------------------------------------------------
extern "C" void kernel_launch(void* const* d_in, const int* in_sizes, int n_in,
                              void* d_out, int out_size, void* d_ws, size_t ws_size,
                              hipStream_t stream) {
  const float* x    = (const float*)d_in[0];  // [4, 4096, 1024]
  const float* Whg  = (const float*)d_in[1];  // [3072, 1024]
  const float* Wout = (const float*)d_in[2];  // [1024, 1536]
  float* out = (float*)d_out;                 // [4, 4096, 1024]

  const size_t HG_ELEMS  = (size_t)MROWS * NCOLS;   // 50,331,648 (192 MB)
  const size_t SB_ELEMS  = (size_t)MROWS * DINNER;  // 25,165,824 ( 96 MB)
  const size_t SUM_ELEMS = (size_t)2 * BATCH * NCHUNK * DINNER;  // 196,608

  float* hg    = (float*)d_ws;
  float* sb    = hg + HG_ELEMS;
  float* sumC  = sb + SB_ELEMS;
  float* sumH  = sumC + SUM_ELEMS;
  float* carry = sumH + SUM_ELEMS;

  const dim3 blk(256);

  // 1) hg = x @ Whg^T  : M=16384, N=3072, K=1024
  gemm_f32_wmma<<<dim3(NCOLS / 128, MROWS / 128), blk, 0, stream>>>(
      x, Whg, hg, MROWS, NCOLS, DIM_);

  // 2) bidirectional chunked linear scan
  scan_partials<<<dim3(DINNER / 256, NCHUNK, 2 * BATCH), blk, 0, stream>>>(hg, sumC, sumH);
  scan_carry<<<dim3((2 * BATCH * DINNER) / 256), blk, 0, stream>>>(sumC, sumH, carry);
  scan_apply<<<dim3(DINNER / 256, NCHUNK, BATCH), blk, 0, stream>>>(hg, carry, sb);

  // 3) out = (scan_f + scan_b) @ Wout^T : M=16384, N=1024, K=1536
  gemm_f32_wmma<<<dim3(DIM_ / 128, MROWS / 128), blk, 0, stream>>>(
      sb, Wout, out, MROWS, DIM_, DINNER);
}